// Attentive_Decoder_22746146799806
// MI455X (gfx1250) — compile-verified
//
#include <hip/hip_runtime.h>
#include <hip/hip_bf16.h>
#include <math.h>

// Problem constants (match reference setup_inputs)
#define BB   128   // batch
#define KK   128   // num objects
#define VD   128   // vote feature dim
#define OD   256   // ref obj feature dim
#define EE   300   // embedding dim
#define AA   512   // attention dim
#define DD   512   // hidden dim
#define VV   4000  // vocab
#define TT   32    // timesteps
#define XD   (EE + VD + OD)   // 684, LSTM input dim
#define ENCD (VD + OD)        // 384

typedef __attribute__((ext_vector_type(2))) float v2f;
typedef __attribute__((ext_vector_type(8))) float v8f;

__device__ __forceinline__ float sigm(float x) { return 1.0f / (1.0f + expf(-x)); }

// ---------------------------------------------------------------------------
// Object mask: objectness softmax + top-16 closest distance mask
// grid = B blocks, 128 threads
// ---------------------------------------------------------------------------
__global__ __launch_bounds__(128) void mask_kernel(
    const float* __restrict__ obj_scores,   // [B,K,2]
    const float* __restrict__ ref_center,   // [B,3]
    const float* __restrict__ xyz,          // [B,K,3]
    float* __restrict__ maskf,              // ws [B,K] 0/1
    float* __restrict__ mask_out)           // d_out tail [B,K]
{
  int b = blockIdx.x, k = threadIdx.x;
  __shared__ float work_s[KK];
  __shared__ float maxd_s;
  const float FINF = __builtin_inff();

  size_t bk = (size_t)b * KK + k;
  float s0 = obj_scores[bk * 2 + 0];
  float s1 = obj_scores[bk * 2 + 1];
  float p = 1.0f / (1.0f + expf(s0 - s1));   // softmax over 2, last element
  int om = p > 0.75f;

  float dx = ref_center[b * 3 + 0] - xyz[bk * 3 + 0];
  float dy = ref_center[b * 3 + 1] - xyz[bk * 3 + 1];
  float dz = ref_center[b * 3 + 2] - xyz[bk * 3 + 2];
  float d  = sqrtf(dx * dx + dy * dy + dz * dz);
  float dm = om ? d : FINF;

  work_s[k] = dm;
  __syncthreads();
  if (k == 0) {
    float last = FINF;
    for (int it = 0; it < 16; ++it) {          // 16th-smallest via selection
      float mn = FINF; int mi = -1;
      for (int j = 0; j < KK; ++j)
        if (work_s[j] < mn) { mn = work_s[j]; mi = j; }
      if (mi >= 0) { work_s[mi] = FINF; last = mn; }
      else { last = FINF; break; }             // <16 finite -> max_d = inf
    }
    maxd_s = last;
  }
  __syncthreads();
  float m = (om && dm <= maxd_s) ? 1.0f : 0.0f;
  maskf[bk] = m;
  mask_out[bk] = m;
}

// ---------------------------------------------------------------------------
// h0/c0 init + copy ref_obj into x-buffer tail. grid = B, 512 threads
// ---------------------------------------------------------------------------
__global__ __launch_bounds__(512) void init_state(
    const float* __restrict__ enc,       // [B,K,VD]
    const float* __restrict__ ref_obj,   // [B,OD]
    const float* __restrict__ W_init_h, const float* __restrict__ b_init_h,
    const float* __restrict__ W_init_c, const float* __restrict__ b_init_c,
    float* __restrict__ h, float* __restrict__ c,
    float* __restrict__ xbuf)            // [B,XD]
{
  int b = blockIdx.x, tid = threadIdx.x;
  __shared__ float in_s[ENCD];
  if (tid < VD) {
    float s = 0.f;
    const float* e0 = enc + (size_t)b * KK * VD + tid;
    for (int k = 0; k < KK; ++k) s += e0[(size_t)k * VD];
    in_s[tid] = s * (1.0f / (float)KK);
  } else if (tid < ENCD) {
    in_s[tid] = ref_obj[(size_t)b * OD + (tid - VD)];
  }
  if (tid < OD)
    xbuf[(size_t)b * XD + (EE + VD) + tid] = ref_obj[(size_t)b * OD + tid];
  __syncthreads();
  const float* wh = W_init_h + (size_t)tid * ENCD;
  const float* wc = W_init_c + (size_t)tid * ENCD;
  float sh = b_init_h[tid], sc = b_init_c[tid];
  for (int j = 0; j < ENCD; ++j) { float x = in_s[j]; sh += x * wh[j]; sc += x * wc[j]; }
  h[(size_t)b * DD + tid] = sh;
  c[(size_t)b * DD + tid] = sc;
}

// ---------------------------------------------------------------------------
// FP32 WMMA GEMM: out[M, N] (+)= A[M,Kd] @ W[N,Kd]^T (+ bias + bias2)
// One wave computes a 16 x (16*NT) output strip. Software-pipelined:
// fragments for k-step i+1 are loaded before the WMMA group of step i, so
// the v_wmma group issues against resident registers while the next loads
// are in flight (one loadcnt wait per group instead of per WMMA).
// A fragment: lane L -> row (L&15), {K=2*(L>>4), K=2*(L>>4)+1}
// B fragment: lane L -> col (L&15), same K split (mirrors A).
// ---------------------------------------------------------------------------
template <int NT>
__global__ __launch_bounds__(32) void wmma_gemm_xt(
    const float* __restrict__ A, const float* __restrict__ W,
    const float* __restrict__ bias, const float* __restrict__ bias2,
    float* __restrict__ out,
    int Kd, int lda, int ldw, int ldo, int accumulate)
{
  int n0 = blockIdx.x * (16 * NT);
  int m0 = blockIdx.y * 16;
  int lane = threadIdx.x;
  int r = lane & 15;
  int half = lane >> 4;

  const float* arow = A + (size_t)(m0 + r) * lda + 2 * half;
  const float* wrow[NT];
#pragma unroll
  for (int j = 0; j < NT; ++j)
    wrow[j] = W + (size_t)(n0 + 16 * j + r) * ldw + 2 * half;

  v8f acc[NT];
#pragma unroll
  for (int j = 0; j < NT; ++j) acc[j] = (v8f){};

  // prologue: fragments for k = 0
  v2f a_c; a_c.x = arow[0]; a_c.y = arow[1];
  v2f b_c[NT];
#pragma unroll
  for (int j = 0; j < NT; ++j) { b_c[j].x = wrow[j][0]; b_c[j].y = wrow[j][1]; }

  for (int k = 4; k < Kd; k += 4) {
    // issue next group's loads first, then compute on the current group
    v2f a_n; a_n.x = arow[k]; a_n.y = arow[k + 1];
    v2f b_n[NT];
#pragma unroll
    for (int j = 0; j < NT; ++j) { b_n[j].x = wrow[j][k]; b_n[j].y = wrow[j][k + 1]; }
#pragma unroll
    for (int j = 0; j < NT; ++j)
      acc[j] = __builtin_amdgcn_wmma_f32_16x16x4_f32(
          false, a_c, false, b_c[j], (short)0, acc[j], false, false);
    a_c = a_n;
#pragma unroll
    for (int j = 0; j < NT; ++j) b_c[j] = b_n[j];
  }
  // epilogue
#pragma unroll
  for (int j = 0; j < NT; ++j)
    acc[j] = __builtin_amdgcn_wmma_f32_16x16x4_f32(
        false, a_c, false, b_c[j], (short)0, acc[j], false, false);

#pragma unroll
  for (int j = 0; j < NT; ++j) {
    int nn = n0 + 16 * j + r;
    float badd = (bias ? bias[nn] : 0.0f) + (bias2 ? bias2[nn] : 0.0f);
#pragma unroll
    for (int e = 0; e < 8; ++e) {
      int mm = m0 + e + 8 * half;
      size_t o = (size_t)mm * ldo + nn;
      float v = acc[j][e] + badd;
      if (accumulate) v += out[o];
      out[o] = v;
    }
  }
}

// ---------------------------------------------------------------------------
// FP32 WMMA GEMM for predictions: preds[b,v,t] = active ? h@W_fc^T + b_fc : 0
// Same software-pipelined 16 x (16*NT) strip; strided store into [B,V,T].
// ---------------------------------------------------------------------------
template <int NT>
__global__ __launch_bounds__(32) void wmma_preds(
    const float* __restrict__ h, const float* __restrict__ W_fc,
    const float* __restrict__ b_fc, float* __restrict__ preds_out,
    const int* __restrict__ lang_len, int t)
{
  int n0 = blockIdx.x * (16 * NT);   // vocab strip
  int m0 = blockIdx.y * 16;          // batch tile
  int lane = threadIdx.x;
  int r = lane & 15;
  int half = lane >> 4;

  const float* arow = h + (size_t)(m0 + r) * DD + 2 * half;
  const float* wrow[NT];
#pragma unroll
  for (int j = 0; j < NT; ++j)
    wrow[j] = W_fc + (size_t)(n0 + 16 * j + r) * DD + 2 * half;

  v8f acc[NT];
#pragma unroll
  for (int j = 0; j < NT; ++j) acc[j] = (v8f){};

  v2f a_c; a_c.x = arow[0]; a_c.y = arow[1];
  v2f b_c[NT];
#pragma unroll
  for (int j = 0; j < NT; ++j) { b_c[j].x = wrow[j][0]; b_c[j].y = wrow[j][1]; }

  for (int k = 4; k < DD; k += 4) {
    v2f a_n; a_n.x = arow[k]; a_n.y = arow[k + 1];
    v2f b_n[NT];
#pragma unroll
    for (int j = 0; j < NT; ++j) { b_n[j].x = wrow[j][k]; b_n[j].y = wrow[j][k + 1]; }
#pragma unroll
    for (int j = 0; j < NT; ++j)
      acc[j] = __builtin_amdgcn_wmma_f32_16x16x4_f32(
          false, a_c, false, b_c[j], (short)0, acc[j], false, false);
    a_c = a_n;
#pragma unroll
    for (int j = 0; j < NT; ++j) b_c[j] = b_n[j];
  }
#pragma unroll
  for (int j = 0; j < NT; ++j)
    acc[j] = __builtin_amdgcn_wmma_f32_16x16x4_f32(
        false, a_c, false, b_c[j], (short)0, acc[j], false, false);

#pragma unroll
  for (int j = 0; j < NT; ++j) {
    int nn = n0 + 16 * j + r;
    float bb = b_fc[nn];
#pragma unroll
    for (int e = 0; e < 8; ++e) {
      int brow = m0 + e + 8 * half;
      int active = (t < lang_len[brow]) ? 1 : 0;
      float v = active ? (acc[j][e] + bb) : 0.0f;
      preds_out[(size_t)brow * VV * TT + (size_t)nn * TT + t] = v;
    }
  }
}

// ---------------------------------------------------------------------------
// Fused attention step: att2, relu-score, masked softmax, gate, awe, emb copy.
// grid = B blocks, 256 threads.
// ---------------------------------------------------------------------------
__global__ __launch_bounds__(256) void attention_step(
    const float* __restrict__ enc,        // [B,K,VD]
    const float* __restrict__ att1,       // [B,K,A]
    const float* __restrict__ maskf,      // [B,K]
    const float* __restrict__ h,          // [B,D]
    const float* __restrict__ W_dec_att, const float* __restrict__ b_dec_att,
    const float* __restrict__ w_full_att, const float* __restrict__ b_full_att,
    const float* __restrict__ W_fbeta,   const float* __restrict__ b_fbeta,
    const float* __restrict__ emb_table, const float* __restrict__ init_emb,
    const int* __restrict__ lang_indices, const int* __restrict__ lang_len,
    float* __restrict__ xbuf,             // [B,XD]
    float* __restrict__ alphas_out,       // [B,T,K]
    int t)
{
  int b = blockIdx.x, tid = threadIdx.x;
  __shared__ float h_s[DD];
  __shared__ float att2_s[AA];
  __shared__ float att_s[KK];
  __shared__ float alpha_s[KK];

  for (int i = tid; i < DD; i += 256) h_s[i] = h[(size_t)b * DD + i];
  __syncthreads();

  // att2 = h @ W_dec_att^T + b
  for (int a = tid; a < AA; a += 256) {
    const float* wr = W_dec_att + (size_t)a * DD;
    float s = b_dec_att[a];
    for (int j = 0; j < DD; ++j) s += h_s[j] * wr[j];
    att2_s[a] = s;
  }
  __syncthreads();

  // att[k] = relu(att1 + att2) . w_full_att + b_full_att
  if (tid < KK) {
    const float* a1 = att1 + ((size_t)b * KK + tid) * AA;
    float s = 0.f;
    for (int a = 0; a < AA; ++a) {
      float v = a1[a] + att2_s[a];
      v = v > 0.f ? v : 0.f;
      s += v * w_full_att[a];
    }
    att_s[tid] = s + b_full_att[0];
  }
  __syncthreads();

  // masked softmax over K (serial: K=128, negligible cost)
  if (tid == 0) {
    const float* mrow = maskf + (size_t)b * KK;
    float m = -__builtin_inff();
    for (int k = 0; k < KK; ++k)
      if (mrow[k] > 0.5f && att_s[k] > m) m = att_s[k];
    if (!(m > -__builtin_inff())) m = 0.0f;    // empty mask -> m = 0
    float dsum = 0.f;
    for (int k = 0; k < KK; ++k) {
      float e = (mrow[k] > 0.5f) ? expf(att_s[k] - m) : 0.0f;
      alpha_s[k] = e;
      dsum += e;
    }
    float inv = (dsum > 0.f) ? 1.0f / fmaxf(dsum, 1e-30f) : 0.0f;
    for (int k = 0; k < KK; ++k) alpha_s[k] *= inv;
  }
  __syncthreads();

  int active = (t < lang_len[b]) ? 1 : 0;
  if (tid < KK)
    alphas_out[((size_t)b * TT + t) * KK + tid] = active ? alpha_s[tid] : 0.0f;

  // gated attention-weighted encoding -> xbuf[EE : EE+VD]
  if (tid < VD) {
    const float* wf = W_fbeta + (size_t)tid * DD;
    float g = b_fbeta[tid];
    for (int j = 0; j < DD; ++j) g += h_s[j] * wf[j];
    g = sigm(g);
    float aw = 0.f;
    const float* e0 = enc + (size_t)b * KK * VD + tid;
    for (int k = 0; k < KK; ++k) aw += alpha_s[k] * e0[(size_t)k * VD];
    xbuf[(size_t)b * XD + EE + tid] = g * aw;
  }

  // teacher-forcing embedding -> xbuf[0:EE]
  const float* esrc = (t == 0)
      ? init_emb
      : (emb_table + (size_t)lang_indices[b * TT + (t - 1)] * EE);
  for (int i = tid; i < EE; i += 256)
    xbuf[(size_t)b * XD + i] = esrc[i];
}

// ---------------------------------------------------------------------------
// LSTM pointwise update (torch gate order i,f,g,o); only active rows update.
// ---------------------------------------------------------------------------
__global__ __launch_bounds__(256) void lstm_pointwise(
    const float* __restrict__ gates,   // [B,4D]
    float* __restrict__ h, float* __restrict__ c,
    const int* __restrict__ lang_len, int t)
{
  int idx = blockIdx.x * 256 + threadIdx.x;   // B*D
  int b = idx >> 9;
  int d = idx & (DD - 1);
  if (t >= lang_len[b]) return;
  const float* g = gates + (size_t)b * (4 * DD);
  float gi = g[d], gf = g[DD + d], gg = g[2 * DD + d], go = g[3 * DD + d];
  float cn = sigm(gf) * c[idx] + sigm(gi) * tanhf(gg);
  float hn = sigm(go) * tanhf(cn);
  c[idx] = cn;
  h[idx] = hn;
}

// ---------------------------------------------------------------------------
extern "C" void kernel_launch(void* const* d_in, const int* in_sizes, int n_in,
                              void* d_out, int out_size, void* d_ws, size_t ws_size,
                              hipStream_t stream) {
  const float* enc        = (const float*)d_in[0];   // [B,K,VD]
  const float* ref_obj    = (const float*)d_in[1];   // [B,OD]
  const float* obj_scores = (const float*)d_in[2];   // [B,K,2]
  const float* ref_center = (const float*)d_in[3];   // [B,3]
  const float* xyz        = (const float*)d_in[4];   // [B,K,3]
  const int*   lang_idx   = (const int*)d_in[5];     // [B,T]
  const int*   lang_len   = (const int*)d_in[6];     // [B]
  const float* emb_table  = (const float*)d_in[7];   // [V,E]
  const float* init_emb   = (const float*)d_in[8];   // [1,E]
  const float* W_enc_att  = (const float*)d_in[9];
  const float* b_enc_att  = (const float*)d_in[10];
  const float* W_dec_att  = (const float*)d_in[11];
  const float* b_dec_att  = (const float*)d_in[12];
  const float* w_full_att = (const float*)d_in[13];
  const float* b_full_att = (const float*)d_in[14];
  const float* W_ih       = (const float*)d_in[15];  // [4D, XD]
  const float* b_ih       = (const float*)d_in[16];
  const float* W_hh       = (const float*)d_in[17];  // [4D, D]
  const float* b_hh       = (const float*)d_in[18];
  const float* W_init_h   = (const float*)d_in[19];
  const float* b_init_h   = (const float*)d_in[20];
  const float* W_init_c   = (const float*)d_in[21];
  const float* b_init_c   = (const float*)d_in[22];
  const float* W_fbeta    = (const float*)d_in[23];
  const float* b_fbeta    = (const float*)d_in[24];
  const float* W_fc       = (const float*)d_in[25];  // [V, D]
  const float* b_fc       = (const float*)d_in[26];

  // workspace layout (floats); total ~35.5 MB
  float* ws    = (float*)d_ws;
  float* maskf = ws;                                   // B*K        = 16384
  float* att1  = maskf + (size_t)BB * KK;              // B*K*A      = 8388608
  float* hbuf  = att1 + (size_t)BB * KK * AA;          // B*D        = 65536
  float* cbuf  = hbuf + (size_t)BB * DD;               // B*D        = 65536
  float* xbuf  = cbuf + (size_t)BB * DD;               // B*XD       = 87552
  float* gates = xbuf + (size_t)BB * XD;               // B*4D       = 262144

  float* out        = (float*)d_out;
  float* preds_out  = out;                             // [B,V,T]
  float* alphas_out = out + (size_t)BB * VV * TT;      // [B,T,K]
  float* mask_out   = alphas_out + (size_t)BB * TT * KK; // [B,K]

  mask_kernel<<<BB, KK, 0, stream>>>(obj_scores, ref_center, xyz, maskf, mask_out);
  init_state<<<BB, DD, 0, stream>>>(enc, ref_obj, W_init_h, b_init_h,
                                    W_init_c, b_init_c, hbuf, cbuf, xbuf);

  // att1 = enc @ W_enc_att^T + b_enc_att  :  [B*K, VD] @ [VD, A]
  wmma_gemm_xt<4><<<dim3(AA / 64, (BB * KK) / 16), 32, 0, stream>>>(
      enc, W_enc_att, b_enc_att, nullptr, att1, VD, VD, VD, AA, 0);

  for (int t = 0; t < TT; ++t) {
    attention_step<<<BB, 256, 0, stream>>>(
        enc, att1, maskf, hbuf, W_dec_att, b_dec_att, w_full_att, b_full_att,
        W_fbeta, b_fbeta, emb_table, init_emb, lang_idx, lang_len,
        xbuf, alphas_out, t);

    // gates = x @ W_ih^T + b_ih + b_hh
    wmma_gemm_xt<4><<<dim3((4 * DD) / 64, BB / 16), 32, 0, stream>>>(
        xbuf, W_ih, b_ih, b_hh, gates, XD, XD, XD, 4 * DD, 0);
    // gates += h @ W_hh^T
    wmma_gemm_xt<4><<<dim3((4 * DD) / 64, BB / 16), 32, 0, stream>>>(
        hbuf, W_hh, nullptr, nullptr, gates, DD, DD, DD, 4 * DD, 1);

    lstm_pointwise<<<(BB * DD) / 256, 256, 0, stream>>>(gates, hbuf, cbuf, lang_len, t);

    // preds[:, :, t] = active ? h @ W_fc^T + b_fc : 0   (4000 = 125 * 32)
    wmma_preds<2><<<dim3(VV / 32, BB / 16), 32, 0, stream>>>(
        hbuf, W_fc, b_fc, preds_out, lang_len, t);
  }
}